// MatrixComplete_17386027614331
// MI455X (gfx1250) — compile-verified
//
#include <hip/hip_runtime.h>

typedef __attribute__((ext_vector_type(2))) float v2f;
typedef __attribute__((ext_vector_type(8))) float v8f;

#define RANK 64

// ---------------------------------------------------------------------------
// Tiled transpose: in [RANK][dim] (row-major) -> out [dim][RANK].
// Block = 256 threads, tile = 64 rows x 64 columns.
// LDS tile stride 65 -> conflict-free stores (stride 65 mod 64 = 1).
// ---------------------------------------------------------------------------
__global__ __launch_bounds__(256) void transpose_rank64(
    const float* __restrict__ in, float* __restrict__ out, int dim) {
  __shared__ float tile[64][65];
  const int t  = threadIdx.x;
  const int c0 = blockIdx.x * 64;
  const int cx = t & 63;        // column within tile
  const int r0 = t >> 6;        // 0..3
  const int c  = c0 + cx;

  #pragma unroll
  for (int k = 0; k < 16; ++k) {
    int r = r0 + (k << 2);      // 0..63
    float v = (c < dim) ? in[(size_t)r * dim + c] : 0.0f;
    tile[cx][r] = v;            // transposed store into LDS
  }
  __syncthreads();

  // out rows c0..c0+63 are contiguous: write as float4 segments.
  #pragma unroll
  for (int k = 0; k < 4; ++k) {
    int f = t + (k << 8);       // flat float4 index 0..1023
    int i = f >> 4;             // tile column == out row offset (0..63)
    int q = f & 15;             // which float4 of the 64-float row
    int row = c0 + i;
    if (row < dim) {
      float4 val;
      val.x = tile[i][q * 4 + 0];
      val.y = tile[i][q * 4 + 1];
      val.z = tile[i][q * 4 + 2];
      val.w = tile[i][q * 4 + 3];
      *(float4*)(out + (size_t)row * RANK + q * 4) = val;
    }
  }
}

// ---------------------------------------------------------------------------
// 16 batch elements per wave via chained V_WMMA_F32_16X16X4_F32.
// A-layout (16x4 f32): VGPR0 = K{0|2} for lanes {0-15|16-31}, VGPR1 = K{1|3}.
// Lane l loads float2 from row i1[b0 + (l&15)] at rank offset 4k + 2*(l>>4),
// which lands directly in A's (and mirrored B's) register layout.
// Diagonal of D = the 16 dot products; biases pre-seeded into C's diagonal.
// ---------------------------------------------------------------------------
__global__ __launch_bounds__(256) void fm_dot_wmma(
    const int* __restrict__ x,
    const float* __restrict__ Ut, const float* __restrict__ Vt,
    const float* __restrict__ bias_U, const float* __restrict__ bias_V,
    float* __restrict__ out, int batch) {
  const int lane = threadIdx.x & 31;
  const int wave = blockIdx.x * (blockDim.x >> 5) + (threadIdx.x >> 5);
  const int b0   = wave << 4;                 // wave-uniform
  if (b0 >= batch) return;                    // uniform exit keeps EXEC all-1s

  const int  m16  = lane & 15;                // which of the 16 batch elems
  const int  half = lane >> 4;                // K-pair selector
  const int  b    = b0 + m16;
  const bool valid = (b < batch);
  const int  bc   = valid ? b : (batch - 1);  // clamp: keep EXEC full

  const int i1 = x[2 * bc];
  const int i2 = x[2 * bc + 1];
  const float bs = bias_U[i1] + bias_V[i2];

  const float* pu = Ut + (size_t)i1 * RANK + (half << 1);
  const float* pv = Vt + (size_t)i2 * RANK + (half << 1);

  // Diagonal (m,m) of D lives at: VGPR m, lane m (m<8) and VGPR m-8, lane m+16.
  const int  sel      = lane & 7;
  const bool diagLane = (lane < 8) || (lane >= 24);

  v8f c;
  #pragma unroll
  for (int d = 0; d < 8; ++d)
    c[d] = (diagLane && sel == d && valid) ? bs : 0.0f;

  #pragma unroll
  for (int k = 0; k < 16; ++k) {
    v2f a  = *(const v2f*)(pu + 4 * k);
    v2f bb = *(const v2f*)(pv + 4 * k);
    c = __builtin_amdgcn_wmma_f32_16x16x4_f32(
        /*neg_a=*/false, a, /*neg_b=*/false, bb,
        /*c_mod=*/(short)0, c, /*reuse_a=*/false, /*reuse_b=*/false);
  }

  if (diagLane) {
    float diag = c[0];
    #pragma unroll
    for (int d = 1; d < 8; ++d) diag = (sel == d) ? c[d] : diag;
    int m  = (lane < 8) ? lane : (lane - 16);
    int ob = b0 + m;
    if (ob < batch) out[ob] = diag;
  }
}

// ---------------------------------------------------------------------------
// Fallback (workspace too small for transposed tables): strided column dot.
// ---------------------------------------------------------------------------
__global__ __launch_bounds__(256) void fm_dot_direct(
    const int* __restrict__ x,
    const float* __restrict__ U_w, const float* __restrict__ V_w,
    const float* __restrict__ bias_U, const float* __restrict__ bias_V,
    float* __restrict__ out, int batch, int dim1, int dim2) {
  int b = blockIdx.x * blockDim.x + threadIdx.x;
  if (b >= batch) return;
  int i1 = x[2 * b];
  int i2 = x[2 * b + 1];
  float acc = 0.0f;
  #pragma unroll 8
  for (int r = 0; r < RANK; ++r)
    acc += U_w[(size_t)r * dim1 + i1] * V_w[(size_t)r * dim2 + i2];
  out[b] = acc + bias_U[i1] + bias_V[i2];
}

extern "C" void kernel_launch(void* const* d_in, const int* in_sizes, int n_in,
                              void* d_out, int out_size, void* d_ws, size_t ws_size,
                              hipStream_t stream) {
  const int*   x      = (const int*)d_in[0];
  const float* U_w    = (const float*)d_in[1];
  const float* V_w    = (const float*)d_in[2];
  const float* bias_U = (const float*)d_in[3];
  const float* bias_V = (const float*)d_in[4];
  float*       out    = (float*)d_out;

  const int batch = in_sizes[0] / 2;
  const int dim1  = in_sizes[1] / RANK;
  const int dim2  = in_sizes[2] / RANK;

  const size_t need = ((size_t)dim1 + (size_t)dim2) * RANK * sizeof(float);
  if (ws_size >= need) {
    float* Ut = (float*)d_ws;
    float* Vt = Ut + (size_t)dim1 * RANK;
    transpose_rank64<<<(dim1 + 63) / 64, 256, 0, stream>>>(U_w, Ut, dim1);
    transpose_rank64<<<(dim2 + 63) / 64, 256, 0, stream>>>(V_w, Vt, dim2);

    const int waves  = (batch + 15) / 16;
    const int blocks = (waves + 7) / 8;   // 8 waves per 256-thread block
    fm_dot_wmma<<<blocks, 256, 0, stream>>>(x, Ut, Vt, bias_U, bias_V, out, batch);
  } else {
    fm_dot_direct<<<(batch + 255) / 256, 256, 0, stream>>>(
        x, U_w, V_w, bias_U, bias_V, out, batch, dim1, dim2);
  }
}